// HeteroGraphODEFunc_73048803770858
// MI455X (gfx1250) — compile-verified
//
#include <hip/hip_runtime.h>

#define NA 50000
#define NT 25000
#define DIN 32
#define HD 64
#define E_AA 800000
#define E_AT 400000
#define E_TA 400000

typedef __attribute__((ext_vector_type(2))) float v2f;
typedef __attribute__((ext_vector_type(8))) float v8f;

static inline int cdiv(int a, int b) { return (a + b - 1) / b; }

__device__ __forceinline__ void atomAddF(float* p, float v) {
    // native global_atomic_add_f32 (no CAS loop)
    unsafeAtomicAdd(p, v);
}

// ---------------- utility kernels ----------------

__global__ void k_zero(float* __restrict__ p, int n) {
    int i = blockIdx.x * blockDim.x + threadIdx.x;
    if (i < n) p[i] = 0.0f;
}

__global__ void k_count(const int* __restrict__ idx, int n, float* __restrict__ deg) {
    int i = blockIdx.x * blockDim.x + threadIdx.x;
    if (i < n) atomAddF(&deg[idx[i]], 1.0f);
}

// mode 0: homo (self-loop: rsqrt(deg+1)); mode 1: bipartite (zero-guarded rsqrt)
__global__ void k_deg2dis(float* __restrict__ d, int n, int mode) {
    int i = blockIdx.x * blockDim.x + threadIdx.x;
    if (i < n) {
        float c = d[i];
        d[i] = (mode == 0) ? rsqrtf(c + 1.0f) : (c > 0.0f ? rsqrtf(c) : 0.0f);
    }
}

// ---------------- WMMA fp32 GEMM: out[M x 64] = act(X[M x K]) @ W[K x 64] -----------
// One wave owns a 16x64 tile (4 C fragments), K fully unrolled in steps of 4 using
// V_WMMA_F32_16X16X4_F32 (exact fp32 semantics). OOB rows are clamped (their D rows
// are never stored), so EXEC stays all-1s with no predicated loads.

template <int K, bool RELU>
__global__ void k_gemm_wmma(const float* __restrict__ X, const float* __restrict__ Wm,
                            float* __restrict__ out, int M) {
    const int lane = threadIdx.x & 31;
    const int wave = threadIdx.x >> 5;
    const int rowBase = (blockIdx.x * (blockDim.x >> 5) + wave) * 16;
    if (rowBase >= M) return;                 // wave-uniform exit

    const int half = lane >> 4;               // 0: K-pair {k0,k0+1}, 1: {k0+2,k0+3}
    const int l    = lane & 15;
    int m = rowBase + l;
    if (m > M - 1) m = M - 1;                 // clamp: garbage lands only in unstored rows
    const float* xrow = X + (size_t)m * K;

    v8f acc0 = {}, acc1 = {}, acc2 = {}, acc3 = {};

#pragma unroll
    for (int k0 = 0; k0 < K; k0 += 4) {
        const int ka = k0 + half * 2;
        float2 a2 = *reinterpret_cast<const float2*>(xrow + ka);   // 8B-aligned
        if (RELU) { a2.x = fmaxf(a2.x, 0.0f); a2.y = fmaxf(a2.y, 0.0f); }
        v2f A; A.x = a2.x; A.y = a2.y;

        const float* wr0 = Wm + (size_t)ka * HD;        // W row ka
        const float* wr1 = Wm + (size_t)(ka + 1) * HD;  // W row ka+1
        v2f B;
        B.x = wr0[l];      B.y = wr1[l];
        acc0 = __builtin_amdgcn_wmma_f32_16x16x4_f32(false, A, false, B, (short)0, acc0, false, false);
        B.x = wr0[16 + l]; B.y = wr1[16 + l];
        acc1 = __builtin_amdgcn_wmma_f32_16x16x4_f32(false, A, false, B, (short)0, acc1, false, false);
        B.x = wr0[32 + l]; B.y = wr1[32 + l];
        acc2 = __builtin_amdgcn_wmma_f32_16x16x4_f32(false, A, false, B, (short)0, acc2, false, false);
        B.x = wr0[48 + l]; B.y = wr1[48 + l];
        acc3 = __builtin_amdgcn_wmma_f32_16x16x4_f32(false, A, false, B, (short)0, acc3, false, false);
    }

    // D layout: VGPR r, lanes 0-15 -> row rowBase+r; lanes 16-31 -> row rowBase+8+r
    const int mlo = rowBase + half * 8;
#pragma unroll
    for (int r = 0; r < 8; ++r) {
        const int mm = mlo + r;
        if (mm < M) {
            float* row = out + (size_t)mm * HD + l;
            row[0]  = acc0[r];
            row[16] = acc1[r];
            row[32] = acc2[r];
            row[48] = acc3[r];
        }
    }
}

// ---------------- accumulator init: acc = (hself * dis^2) + b0 (+ b1) ----------------

__global__ void k_init_acc(const float* __restrict__ hself, const float* __restrict__ dis,
                           const float* __restrict__ b0, const float* __restrict__ b1,
                           float* __restrict__ acc, int n) {
    int i = blockIdx.x * blockDim.x + threadIdx.x;
    if (i < n * HD) {
        int node = i >> 6, j = i & 63;
        float v = b0[j];
        if (b1) v += b1[j];
        if (hself) { float s = dis[node]; v += hself[i] * s * s; }
        acc[i] = v;
    }
}

// ---------------- edge scatter: acc[dst] += h[src] * dis_s[src] * dis_d[dst] --------
// 16 threads per edge, float4 gather per thread, native f32 atomics (L2-resident).

__global__ void k_scatter(const float* __restrict__ h, const int* __restrict__ src,
                          const int* __restrict__ dst, const float* __restrict__ dis_s,
                          const float* __restrict__ dis_d, float* __restrict__ acc, int E) {
    int t = blockIdx.x * blockDim.x + threadIdx.x;
    int e = t >> 4;
    if (e >= E) return;
    int q = (t & 15) << 2;
    int s = src[e], d = dst[e];
    float c = dis_s[s] * dis_d[d];
    const float4 v = *reinterpret_cast<const float4*>(h + (size_t)s * HD + q);
    float* o = acc + (size_t)d * HD + q;
    atomAddF(o + 0, v.x * c);
    atomAddF(o + 1, v.y * c);
    atomAddF(o + 2, v.z * c);
    atomAddF(o + 3, v.w * c);
}

// ---------------- output projection: out[i,:2] = g[i,:] @ Wp + bp -------------------

__global__ void k_proj(const float* __restrict__ g, const float* __restrict__ Wp,
                       const float* __restrict__ bp, float* __restrict__ out, int n) {
    int i = blockIdx.x * blockDim.x + threadIdx.x;
    if (i < n) {
        float a0 = bp[0], a1 = bp[1];
        const float4* row4 = reinterpret_cast<const float4*>(g + (size_t)i * HD);
        const float4* Wp4  = reinterpret_cast<const float4*>(Wp);
#pragma unroll
        for (int jj = 0; jj < HD / 4; ++jj) {
            float4 g4 = row4[jj];
            float4 w0 = Wp4[2 * jj];      // rows 4jj, 4jj+1
            float4 w1 = Wp4[2 * jj + 1];  // rows 4jj+2, 4jj+3
            a0 += g4.x * w0.x + g4.y * w0.z + g4.z * w1.x + g4.w * w1.z;
            a1 += g4.x * w0.y + g4.y * w0.w + g4.z * w1.y + g4.w * w1.w;
        }
        out[i * 2 + 0] = a0;
        out[i * 2 + 1] = a1;
    }
}

// ---------------- launcher ----------------

extern "C" void kernel_launch(void* const* d_in, const int* in_sizes, int n_in,
                              void* d_out, int out_size, void* d_ws, size_t ws_size,
                              hipStream_t stream) {
    (void)in_sizes; (void)n_in; (void)out_size; (void)ws_size;

    const float* x_agent  = (const float*)d_in[1];
    const float* x_target = (const float*)d_in[2];
    const int* src_aa = (const int*)d_in[3];
    const int* dst_aa = (const int*)d_in[4];
    const int* src_at = (const int*)d_in[5];
    const int* dst_at = (const int*)d_in[6];
    const int* src_ta = (const int*)d_in[7];
    const int* dst_ta = (const int*)d_in[8];
    const float* W1_aa = (const float*)d_in[9];  const float* b1_aa = (const float*)d_in[10];
    const float* W1_at = (const float*)d_in[11]; const float* b1_at = (const float*)d_in[12];
    const float* W1_ta = (const float*)d_in[13]; const float* b1_ta = (const float*)d_in[14];
    const float* W2_aa = (const float*)d_in[15]; const float* b2_aa = (const float*)d_in[16];
    const float* W2_at = (const float*)d_in[17]; const float* b2_at = (const float*)d_in[18];
    const float* W2_ta = (const float*)d_in[19]; const float* b2_ta = (const float*)d_in[20];
    const float* Wp_a  = (const float*)d_in[21]; const float* bp_a  = (const float*)d_in[22];
    const float* Wp_t  = (const float*)d_in[23]; const float* bp_t  = (const float*)d_in[24];

    float* out_agent  = (float*)d_out;            // [NA*2]
    float* out_target = out_agent + 2 * NA;       // [NT*2]

    // workspace layout (floats)
    float* W = (float*)d_ws;
    float* dis_aa   = W;                          // NA  (dst_aa degrees -> dis)
    float* dis_at_s = W + NA;                     // NA
    float* dis_ta_d = W + 2 * (size_t)NA;         // NA
    float* dis_at_d = W + 3 * (size_t)NA;         // NT
    float* dis_ta_s = W + 3 * (size_t)NA + NT;    // NT
    float* h1_aa  = W + 3 * (size_t)NA + 2 * (size_t)NT;  // NA*HD
    float* h1_ta  = h1_aa  + (size_t)NA * HD;             // NT*HD
    float* h1_at  = h1_ta  + (size_t)NT * HD;             // NA*HD
    float* acc_a  = h1_at  + (size_t)NA * HD;             // NA*HD
    float* acc_t  = acc_a  + (size_t)NA * HD;             // NT*HD
    float* acc2_a = acc_t  + (size_t)NT * HD;             // NA*HD
    float* acc2_t = acc2_a + (size_t)NA * HD;             // NT*HD
    // layer-2 GEMM outputs reuse the layer-1 h buffers
    float* g2_aa = h1_aa; float* g2_ta = h1_ta; float* g2_at = h1_at;

    const int B = 256;

    // degrees (re-zeroed every call for determinism)
    int ndeg = 3 * NA + 2 * NT;
    k_zero<<<cdiv(ndeg, B), B, 0, stream>>>(W, ndeg);
    k_count<<<cdiv(E_AA, B), B, 0, stream>>>(dst_aa, E_AA, dis_aa);
    k_count<<<cdiv(E_AT, B), B, 0, stream>>>(src_at, E_AT, dis_at_s);
    k_count<<<cdiv(E_AT, B), B, 0, stream>>>(dst_at, E_AT, dis_at_d);
    k_count<<<cdiv(E_TA, B), B, 0, stream>>>(src_ta, E_TA, dis_ta_s);
    k_count<<<cdiv(E_TA, B), B, 0, stream>>>(dst_ta, E_TA, dis_ta_d);
    k_deg2dis<<<cdiv(NA, B), B, 0, stream>>>(dis_aa, NA, 0);
    k_deg2dis<<<cdiv(NA, B), B, 0, stream>>>(dis_at_s, NA, 1);
    k_deg2dis<<<cdiv(NT, B), B, 0, stream>>>(dis_at_d, NT, 1);
    k_deg2dis<<<cdiv(NT, B), B, 0, stream>>>(dis_ta_s, NT, 1);
    k_deg2dis<<<cdiv(NA, B), B, 0, stream>>>(dis_ta_d, NA, 1);

    // ---- layer 1 GEMMs (K = DIN = 32), 8 waves/block -> 128 rows/block ----
    k_gemm_wmma<DIN, false><<<cdiv(NA, 128), B, 0, stream>>>(x_agent,  W1_aa, h1_aa, NA);
    k_gemm_wmma<DIN, false><<<cdiv(NT, 128), B, 0, stream>>>(x_target, W1_ta, h1_ta, NT);
    k_gemm_wmma<DIN, false><<<cdiv(NA, 128), B, 0, stream>>>(x_agent,  W1_at, h1_at, NA);

    // acc init: agent gets homo self-loop term + both biases; target gets bias only
    k_init_acc<<<cdiv(NA * HD, B), B, 0, stream>>>(h1_aa, dis_aa, b1_aa, b1_ta, acc_a, NA);
    k_init_acc<<<cdiv(NT * HD, B), B, 0, stream>>>(nullptr, nullptr, b1_at, nullptr, acc_t, NT);

    // edge scatter
    k_scatter<<<cdiv(E_AA * 16, B), B, 0, stream>>>(h1_aa, src_aa, dst_aa, dis_aa, dis_aa, acc_a, E_AA);
    k_scatter<<<cdiv(E_TA * 16, B), B, 0, stream>>>(h1_ta, src_ta, dst_ta, dis_ta_s, dis_ta_d, acc_a, E_TA);
    k_scatter<<<cdiv(E_AT * 16, B), B, 0, stream>>>(h1_at, src_at, dst_at, dis_at_s, dis_at_d, acc_t, E_AT);

    // ---- layer 2 GEMMs (K = H = 64), ReLU fused into the A-operand load ----
    k_gemm_wmma<HD, true><<<cdiv(NA, 128), B, 0, stream>>>(acc_a, W2_aa, g2_aa, NA);
    k_gemm_wmma<HD, true><<<cdiv(NT, 128), B, 0, stream>>>(acc_t, W2_ta, g2_ta, NT);
    k_gemm_wmma<HD, true><<<cdiv(NA, 128), B, 0, stream>>>(acc_a, W2_at, g2_at, NA);

    k_init_acc<<<cdiv(NA * HD, B), B, 0, stream>>>(g2_aa, dis_aa, b2_aa, b2_ta, acc2_a, NA);
    k_init_acc<<<cdiv(NT * HD, B), B, 0, stream>>>(nullptr, nullptr, b2_at, nullptr, acc2_t, NT);

    k_scatter<<<cdiv(E_AA * 16, B), B, 0, stream>>>(g2_aa, src_aa, dst_aa, dis_aa, dis_aa, acc2_a, E_AA);
    k_scatter<<<cdiv(E_TA * 16, B), B, 0, stream>>>(g2_ta, src_ta, dst_ta, dis_ta_s, dis_ta_d, acc2_a, E_TA);
    k_scatter<<<cdiv(E_AT * 16, B), B, 0, stream>>>(g2_at, src_at, dst_at, dis_at_s, dis_at_d, acc2_t, E_AT);

    // ---- output projections ----
    k_proj<<<cdiv(NA, B), B, 0, stream>>>(acc2_a, Wp_a, bp_a, out_agent, NA);
    k_proj<<<cdiv(NT, B), B, 0, stream>>>(acc2_t, Wp_t, bp_t, out_target, NT);
}